// SoftmaxAttentionModel_4183298146983
// MI455X (gfx1250) — compile-verified
//
#include <hip/hip_runtime.h>

typedef float v2f __attribute__((ext_vector_type(2)));
typedef float v8f __attribute__((ext_vector_type(8)));

#define B_N   4096
#define L_N   128
#define D_N   64
#define NEGINF (-1.0e9f)
#define COS_EPS 1e-8f

__global__ __launch_bounds__(256) void cos_attn_kernel(
    const int* __restrict__ ids_a,  const int* __restrict__ mask_a,
    const int* __restrict__ ids_b,  const int* __restrict__ mask_b,
    const float* __restrict__ mu_table, const float* __restrict__ feat_table,
    float* __restrict__ out)
{
    __shared__ __align__(16) float tile[L_N * D_N];  // 32 KB: mu rows, then reused for feat
    __shared__ float q[D_N];
    __shared__ float attn[L_N];           // scores, then attention weights
    __shared__ int   s_ids[L_N];
    __shared__ float s_mask[L_N];
    __shared__ float partial[2][D_N];
    __shared__ float sideout[2][D_N];
    __shared__ float red[3 * D_N];

    const int t    = threadIdx.x;
    const int b    = blockIdx.x;
    const int wave = t >> 5;
    const int lane = t & 31;
    const int half = (lane >> 4) & 1;   // which K-half of the 16x4 A tile this lane feeds
    const int mrow = lane & 15;

    float4* __restrict__ tile4 = reinterpret_cast<float4*>(tile);
    const float4* __restrict__ mu4   = reinterpret_cast<const float4*>(mu_table);
    const float4* __restrict__ feat4 = reinterpret_cast<const float4*>(feat_table);

    for (int side = 0; side < 2; ++side) {
        const int* ids  = (side == 0) ? ids_a  : ids_b;
        const int* mask = (side == 0) ? mask_a : mask_b;

        if (t < L_N) {
            s_ids[t]  = ids [(size_t)b * L_N + t];
            s_mask[t] = (float)mask[(size_t)b * L_N + t];
        }
        __syncthreads();

        // ---- stage gathered mu rows into LDS: b128 loads, 16B/lane ----
        #pragma unroll
        for (int i = t; i < L_N * (D_N / 4); i += 256) {   // 2048 float4s
            int l = i >> 4, d4 = i & 15;
            tile4[i] = mu4[(size_t)s_ids[l] * (D_N / 4) + d4];
        }
        __syncthreads();

        // ---- masked mean pool -> q[64] ----
        if (t < D_N) {
            float acc = 0.f, msum = 0.f;
            for (int l = 0; l < L_N; ++l) {
                float m = s_mask[l];
                acc  += m * tile[l * D_N + t];
                msum += m;
            }
            q[t] = acc / fmaxf(msum, 1.0f);
        }
        __syncthreads();

        // ---- scores = mu(128x64) @ q(64) via V_WMMA_F32_16X16X4_F32 ----
        {
            const int lbase = wave * 16;             // 8 waves x 16 rows = 128 rows
            v8f c = {0.f,0.f,0.f,0.f,0.f,0.f,0.f,0.f};
            #pragma unroll
            for (int kk = 0; kk < 16; ++kk) {        // 16 chunks of K=4 cover D=64
                int k0 = kk * 4 + half * 2;
                v2f a, bq;
                a.x  = tile[(lbase + mrow) * D_N + k0 + 0];  // A[m][k]
                a.y  = tile[(lbase + mrow) * D_N + k0 + 1];
                bq.x = q[k0 + 0];                            // B[k][n] = q[k] broadcast
                bq.y = q[k0 + 1];
                c = __builtin_amdgcn_wmma_f32_16x16x4_f32(
                        false, a, false, bq, (short)0, c, false, false);
            }
            // every column of D equals the score; lanes 0/16 scatter the 16 rows
            if (lane == 0)
                for (int v = 0; v < 8; ++v) attn[lbase + v] = c[v];
            if (lane == 16)
                for (int v = 0; v < 8; ++v) attn[lbase + 8 + v] = c[v];
        }
        __syncthreads();

        // ---- masked softmax over L=128 (redundant per-thread, broadcast reads) ----
        float mx = NEGINF;
        for (int j = 0; j < L_N; ++j) {
            float s = (s_mask[j] != 0.f) ? attn[j] : NEGINF;
            mx = fmaxf(mx, s);
        }
        float sum = 0.f;
        for (int j = 0; j < L_N; ++j) {
            float s = (s_mask[j] != 0.f) ? attn[j] : NEGINF;
            sum += expf(s - mx);
        }
        float mine = 0.f;
        if (t < L_N) mine = (s_mask[t] != 0.f) ? attn[t] : NEGINF;
        __syncthreads();
        if (t < L_N) attn[t] = expf(mine - mx) / sum;
        __syncthreads();

        // ---- stage gathered feat rows into LDS (reuse tile): b128 loads ----
        #pragma unroll
        for (int i = t; i < L_N * (D_N / 4); i += 256) {
            int l = i >> 4, d4 = i & 15;
            tile4[i] = feat4[(size_t)s_ids[l] * (D_N / 4) + d4];
        }
        __syncthreads();

        // ---- out = attn(1x128) @ feat(128x64) via WMMA ----
        {
            const int n0 = (wave & 3) * 16;   // column tile
            const int lh = wave >> 2;         // K half: l in [64*lh, 64*lh+64)
            const int nc = lane & 15;
            v8f c = {0.f,0.f,0.f,0.f,0.f,0.f,0.f,0.f};
            #pragma unroll
            for (int kk = 0; kk < 16; ++kk) {
                int kb = lh * 64 + kk * 4 + half * 2;
                v2f a, bf;
                a.x  = attn[kb + 0];                       // A[m][k] = attn broadcast over m
                a.y  = attn[kb + 1];
                bf.x = tile[(kb + 0) * D_N + n0 + nc];     // B[k][n] = feat[l=k][n]
                bf.y = tile[(kb + 1) * D_N + n0 + nc];
                c = __builtin_amdgcn_wmma_f32_16x16x4_f32(
                        false, a, false, bf, (short)0, c, false, false);
            }
            // row 0 of D holds the partial output for this column tile
            if (lane < 16) partial[lh][n0 + lane] = c[0];
        }
        __syncthreads();
        if (t < D_N) sideout[side][t] = partial[0][t] + partial[1][t];
        __syncthreads();
    }

    // ---- cosine similarity * 5 ----
    if (t < D_N) {
        float av = sideout[0][t], bv = sideout[1][t];
        red[t]           = av * bv;
        red[D_N + t]     = av * av;
        red[2 * D_N + t] = bv * bv;
    }
    __syncthreads();
    if (t == 0) {
        float dot = 0.f, na = 0.f, nb = 0.f;
        for (int d = 0; d < D_N; ++d) {
            dot += red[d];
            na  += red[D_N + d];
            nb  += red[2 * D_N + d];
        }
        na = fmaxf(sqrtf(na), COS_EPS);
        nb = fmaxf(sqrtf(nb), COS_EPS);
        out[b] = 5.0f * dot / (na * nb);
    }
}

extern "C" void kernel_launch(void* const* d_in, const int* in_sizes, int n_in,
                              void* d_out, int out_size, void* d_ws, size_t ws_size,
                              hipStream_t stream) {
    (void)in_sizes; (void)n_in; (void)out_size; (void)d_ws; (void)ws_size;
    const int*   ids_a    = (const int*)  d_in[0];
    const int*   mask_a   = (const int*)  d_in[1];
    const int*   ids_b    = (const int*)  d_in[2];
    const int*   mask_b   = (const int*)  d_in[3];
    const float* mu_tab   = (const float*)d_in[4];
    const float* feat_tab = (const float*)d_in[5];
    float*       outp     = (float*)      d_out;

    cos_attn_kernel<<<B_N, 256, 0, stream>>>(ids_a, mask_a, ids_b, mask_b,
                                             mu_tab, feat_tab, outp);
}